// Explore_80092550136107
// MI455X (gfx1250) — compile-verified
//
#include <hip/hip_runtime.h>
#include <math.h>

#define N_NODES_C 200000
#define N_EDGES_C 1000000
#define N_Q_C 64
#define N_REL_C 401
#define D_C 128
#define A_C 64

typedef __attribute__((ext_vector_type(2))) float v2f;
typedef __attribute__((ext_vector_type(8))) float v8f;

__global__ void zero_kernel(float* __restrict__ p, int n) {
    int i = blockIdx.x * blockDim.x + threadIdx.x;
    int stride = gridDim.x * blockDim.x;
    for (; i < n; i += stride) p[i] = 0.0f;
}

// Out[m,n] = act( sum_k X[m,k] * W[n,k] )   (einsum 'mk,nk->mn')
// One wave computes a full 16-row x (COLT*16)-col strip: COLT accumulators,
// A fragment loaded once per K-step and reused across COLT WMMAs.
// Uses V_WMMA_F32_16X16X4_F32 (fp32 in, fp32 acc). Wave-uniform guards only.
template <int COLT, bool RELU>
__global__ __launch_bounds__(256) void gemm_xwt_strip_kernel(
        const float* __restrict__ X,
        const float* __restrict__ W,
        float* __restrict__ Out,
        int M, int K) {
    const int lane  = threadIdx.x & 31;
    const int wave  = threadIdx.x >> 5;
    const int strip = blockIdx.x * 8 + wave;       // wave-uniform
    if (strip * 16 >= M) return;                   // EXEC all-1s below

    const int mn = lane & 15;                      // row (A) / col (B,D) within tile
    const int hi = lane >> 4;                      // K-half selector
    const int Ncols = COLT * 16;

    const float* xrow = X + (size_t)(strip * 16 + mn) * K;
    const float* wrow = W + (size_t)mn * K;        // ct strides added as immediates

    v8f acc[COLT];
    #pragma unroll
    for (int ct = 0; ct < COLT; ++ct) acc[ct] = (v8f){};

    for (int k0 = 0; k0 < K; k0 += 4) {
        // A 16x4 f32: lanes 0-15 hold K=0,1 ; lanes 16-31 hold K=2,3 (ISA 7.12.2)
        v2f a = *(const v2f*)(xrow + k0 + 2 * hi);
        #pragma unroll
        for (int ct = 0; ct < COLT; ++ct) {
            // B 4x16 f32 (B[k][n] = W[n][k]); W row block ct at constant offset
            v2f b = *(const v2f*)(wrow + (size_t)ct * 16 * K + k0 + 2 * hi);
            acc[ct] = __builtin_amdgcn_wmma_f32_16x16x4_f32(
                /*neg_a=*/false, a, /*neg_b=*/false, b,
                /*c_mod=*/(short)0, acc[ct], /*reuse_a=*/false, /*reuse_b=*/false);
        }
    }

    // D 16x16 f32: VGPR r -> row r (lanes 0-15) / row r+8 (lanes 16-31), col = lane&15
    const size_t rowBase = (size_t)(strip * 16 + 8 * hi);
    #pragma unroll
    for (int ct = 0; ct < COLT; ++ct) {
        #pragma unroll
        for (int r = 0; r < 8; ++r) {
            float v = acc[ct][r];
            if (RELU) v = fmaxf(v, 0.0f);
            Out[(rowBase + r) * (size_t)Ncols + ct * 16 + mn] = v;
        }
    }
}

// combo[rel,q,a] = Ws_b[a] + sum_d hr[d]*Wr[a,d] + hq[d]*Wq[a,d] + hr[d]*hq[d]*Wqr[a,d]
__global__ void combo_kernel(const float* __restrict__ q_emb,
                             const float* __restrict__ rela,
                             const float* __restrict__ Wr,
                             const float* __restrict__ Wq,
                             const float* __restrict__ Wqr,
                             const float* __restrict__ Ws_b,
                             float* __restrict__ combo) {
    const int q = blockIdx.x;     // 0..63
    const int rl = blockIdx.y;    // 0..400
    const int a = threadIdx.x;    // 0..63
    __shared__ float shr[D_C];
    __shared__ float shq[D_C];
    shr[a]      = rela[(size_t)rl * D_C + a];
    shr[a + 64] = rela[(size_t)rl * D_C + a + 64];
    shq[a]      = q_emb[(size_t)q * D_C + a];
    shq[a + 64] = q_emb[(size_t)q * D_C + a + 64];
    __syncthreads();

    const float* wr  = Wr  + (size_t)a * D_C;
    const float* wq  = Wq  + (size_t)a * D_C;
    const float* wqr = Wqr + (size_t)a * D_C;
    float s = Ws_b[a];
    #pragma unroll 4
    for (int d = 0; d < D_C; ++d) {
        const float hr = shr[d], hq = shq[d];
        s = fmaf(hr, wr[d], s);
        s = fmaf(hq, wq[d], s);
        s = fmaf(hr * hq, wqr[d], s);
    }
    combo[((size_t)rl * N_Q_C + q) * A_C + a] = s;
}

// One wave (32 lanes) per edge: alpha = sigmoid(relu(proj[sub]+combo[rel,q]) . walpha + b)
// then atomic scatter of alpha*(hs*hr) into agg[obj].
__global__ void edge_kernel(const float* __restrict__ hidden,
                            const float* __restrict__ rela,
                            const float* __restrict__ proj,
                            const float* __restrict__ combo,
                            const float* __restrict__ walpha_w,
                            const float* __restrict__ walpha_b,
                            const int* __restrict__ r_idx,
                            const int* __restrict__ rel,
                            const int* __restrict__ sub,
                            const int* __restrict__ obj,
                            float* __restrict__ agg,
                            float* __restrict__ alpha_out) {
    const int lane = threadIdx.x & 31;
    const int wave = threadIdx.x >> 5;
    const int e = blockIdx.x * 8 + wave;   // wave-uniform
    if (e >= N_EDGES_C) return;

    const int s = sub[e];
    const int r = rel[e];
    const int q = r_idx[e];
    const int o = obj[e];

    const float* pr = proj  + (size_t)s * A_C;
    const float* cb = combo + ((size_t)r * N_Q_C + q) * A_C;
    float p0 = fmaxf(pr[lane]      + cb[lane],      0.0f);
    float p1 = fmaxf(pr[lane + 32] + cb[lane + 32], 0.0f);
    float dot = p0 * walpha_w[lane] + p1 * walpha_w[lane + 32];
    #pragma unroll
    for (int off = 16; off > 0; off >>= 1)
        dot += __shfl_xor(dot, off, 32);

    const float alpha = 1.0f / (1.0f + __expf(-(dot + walpha_b[0])));
    if (lane == 0) alpha_out[e] = alpha;

    const float* hs = hidden + (size_t)s * D_C;
    const float* hr = rela   + (size_t)r * D_C;
    float* ag = agg + (size_t)o * D_C;
    #pragma unroll
    for (int i = 0; i < 4; ++i) {
        const int d = lane + 32 * i;
        atomicAdd(&ag[d], alpha * hs[d] * hr[d]);
    }
}

extern "C" void kernel_launch(void* const* d_in, const int* in_sizes, int n_in,
                              void* d_out, int out_size, void* d_ws, size_t ws_size,
                              hipStream_t stream) {
    const float* hidden   = (const float*)d_in[0];
    const float* q_emb    = (const float*)d_in[1];
    const float* rela     = (const float*)d_in[2];
    const float* Ws_w     = (const float*)d_in[3];
    const float* Ws_b     = (const float*)d_in[4];
    const float* Wr_w     = (const float*)d_in[5];
    const float* Wq_w     = (const float*)d_in[6];
    const float* Wqr_w    = (const float*)d_in[7];
    const float* walpha_w = (const float*)d_in[8];
    const float* walpha_b = (const float*)d_in[9];
    const float* Wh_w     = (const float*)d_in[10];
    const int* r_idx = (const int*)d_in[11];
    const int* rel   = (const int*)d_in[12];
    const int* sub   = (const int*)d_in[13];
    const int* obj   = (const int*)d_in[14];

    float* out = (float*)d_out;                       // [N*D hidden_new | E alpha]
    float* agg   = (float*)d_ws;                      // N*D floats
    float* proj  = agg  + (size_t)N_NODES_C * D_C;    // N*A floats
    float* combo = proj + (size_t)N_NODES_C * A_C;    // 401*64*64 floats

    const int strips = N_NODES_C / 16;                // 12500 16-row strips
    const int stripBlocks = (strips + 7) / 8;         // 8 waves per block

    // 1) zero the aggregation buffer
    zero_kernel<<<2048, 256, 0, stream>>>(agg, N_NODES_C * D_C);

    // 2) node projection: proj = hidden @ Ws_w^T   [200000 x 64], WMMA f32
    gemm_xwt_strip_kernel<4, false><<<stripBlocks, 256, 0, stream>>>(
        hidden, Ws_w, proj, N_NODES_C, D_C);

    // 3) (rel, q) combo table
    {
        dim3 grid(N_Q_C, N_REL_C);
        combo_kernel<<<grid, 64, 0, stream>>>(q_emb, rela, Wr_w, Wq_w, Wqr_w, Ws_b, combo);
    }

    // 4) per-edge alpha + gated message scatter
    edge_kernel<<<(N_EDGES_C + 7) / 8, 256, 0, stream>>>(
        hidden, rela, proj, combo, walpha_w, walpha_b,
        r_idx, rel, sub, obj, agg, out + (size_t)N_NODES_C * D_C);

    // 5) hidden_new = relu(agg @ Wh_w^T)   [200000 x 128], WMMA f32
    gemm_xwt_strip_kernel<8, true><<<stripBlocks, 256, 0, stream>>>(
        agg, Wh_w, out, N_NODES_C, D_C);
}